// DetFun_14474039788027
// MI455X (gfx1250) — compile-verified
//
#include <hip/hip_runtime.h>

// ---------------------------------------------------------------------------
// CDNA5 (gfx1250) block-state IIR cascade via V_WMMA_F32_16X16X4_F32.
//
//  x: [B=16, T=16384, C=64] fp32.  Pipeline: bandpass(2 biquads) -> square ->
//  lowpass(2 biquads), independent per (b,c) channel along T.
//
//  Per 16-sample block, a 2-biquad stage is linear:
//     Y  = Hc(16x16)*X + Pc(16x4)*S
//     S' = Qc(4x16)*X  + Rc(4x4)*S        (S = 4 DF2T states of the cascade)
//  -> dense fp32 WMMAs (16x16x4, K-chained x4 for the 16x16 products).
//
//  T is cut into 32 chunks of 512 (32 blocks each).  Linear-scan decomposition
//  across chunks: sweep1 computes zero-state chunk exit states V (parallel),
//  a tiny scan computes true entry states (S_in[k+1] = Rc^32 * S_in[k] + V[k]),
//  exact re-sweeps then run with correct states.  3 x-sweeps total; x (67 MB)
//  is L2-resident (192 MB L2), so HBM traffic ~ 67 MB read + 67 MB write.
// ---------------------------------------------------------------------------

typedef __attribute__((ext_vector_type(2))) float v2f;
typedef __attribute__((ext_vector_type(8))) float v8f;

#define TLEN   16384
#define CCH    64
#define NCHUNK 32
#define LCH    512
#define NBLK   32      // LCH / 16
#define NCT    64      // channel tiles: 16 batches * 4 tiles of 16 channels

// coefficient block layout (floats), one per stage
#define OFF_HC 0       // Hc   [16][16]
#define OFF_PC 256     // Pc   [16][4]
#define OFF_QP 320     // Qc   padded [16][16] (rows >=4 zero)
#define OFF_RP 576     // Rc   padded [16][4]  (rows >=4 zero)
#define OFF_RT 640     // Rc^NBLK [4][4]
#define COEF_STRIDE 768

#define STATES_FLOATS (NCHUNK * NCT * 64)   // 4 states x 16 ch per (chunk,ctile)
#define OFF_V1  (2 * COEF_STRIDE)
#define OFF_ST1 (OFF_V1  + STATES_FLOATS)
#define OFF_V2  (OFF_ST1 + STATES_FLOATS)
#define OFF_ST2 (OFF_V2  + STATES_FLOATS)

// ---------------------------------------------------------------------------
// Setup: derive block matrices on-device from SOS coefficients (1 lane).
// ---------------------------------------------------------------------------
struct BQ { float b0, b1, b2, a1, a2; };

__device__ void bqSim(const BQ q, const float* x, float s1, float s2,
                      float* y, float* s1o, float* s2o) {
  for (int i = 0; i < 16; ++i) {
    float xi = x ? x[i] : 0.f;
    float yi = q.b0 * xi + s1;
    float n1 = q.b1 * xi - q.a1 * yi + s2;
    float n2 = q.b2 * xi - q.a2 * yi;
    s1 = n1; s2 = n2;
    if (y) y[i] = yi;
  }
  *s1o = s1; *s2o = s2;
}

// H[16][16], P[16][2], Q[2][16], R[2][2] for one biquad over a 16-block
__device__ void bqBlock(const BQ q, float* H, float* P, float* Qm, float* R) {
  float x[16], y[16], s1, s2;
  for (int c = 0; c < 16; ++c) {
    for (int i = 0; i < 16; ++i) x[i] = (i == c) ? 1.f : 0.f;
    bqSim(q, x, 0.f, 0.f, y, &s1, &s2);
    for (int r = 0; r < 16; ++r) H[r * 16 + c] = y[r];
    Qm[0 * 16 + c] = s1; Qm[1 * 16 + c] = s2;
  }
  bqSim(q, nullptr, 1.f, 0.f, y, &s1, &s2);
  for (int r = 0; r < 16; ++r) P[r * 2 + 0] = y[r];
  R[0 * 2 + 0] = s1; R[1 * 2 + 0] = s2;
  bqSim(q, nullptr, 0.f, 1.f, y, &s1, &s2);
  for (int r = 0; r < 16; ++r) P[r * 2 + 1] = y[r];
  R[0 * 2 + 1] = s1; R[1 * 2 + 1] = s2;
}

__device__ void mm(const float* A, const float* B, float* Cm, int M, int K, int N) {
  for (int i = 0; i < M; ++i)
    for (int j = 0; j < N; ++j) {
      float s = 0.f;
      for (int k = 0; k < K; ++k) s += A[i * K + k] * B[k * N + j];
      Cm[i * N + j] = s;
    }
}

__device__ void buildStage(const float* sA, const float* sB, float* w) {
  BQ qa = { sA[0]/sA[3], sA[1]/sA[3], sA[2]/sA[3], sA[4]/sA[3], sA[5]/sA[3] };
  BQ qb = { sB[0]/sB[3], sB[1]/sB[3], sB[2]/sB[3], sB[4]/sB[3], sB[5]/sB[3] };
  float HA[256], PA[32], QA[32], RA[4];
  float HB[256], PB[32], QB[32], RB[4];
  bqBlock(qa, HA, PA, QA, RA);
  bqBlock(qb, HB, PB, QB, RB);

  float Hc[256];  mm(HB, HA, Hc,   16, 16, 16);   // cascade block response
  float HBPA[32]; mm(HB, PA, HBPA, 16, 16, 2);
  float QBHA[32]; mm(QB, HA, QBHA, 2, 16, 16);
  float QBPA[4];  mm(QB, PA, QBPA, 2, 16, 2);

  for (int i = 0; i < 256; ++i) w[OFF_HC + i] = Hc[i];
  for (int r = 0; r < 16; ++r) {                       // Pc = [HB*PA | PB]
    w[OFF_PC + r * 4 + 0] = HBPA[r * 2 + 0];
    w[OFF_PC + r * 4 + 1] = HBPA[r * 2 + 1];
    w[OFF_PC + r * 4 + 2] = PB[r * 2 + 0];
    w[OFF_PC + r * 4 + 3] = PB[r * 2 + 1];
  }
  for (int i = 0; i < 256; ++i) w[OFF_QP + i] = 0.f;   // Qc = [QA ; QB*HA]
  for (int c = 0; c < 16; ++c) {
    w[OFF_QP + 0 * 16 + c] = QA[0 * 16 + c];
    w[OFF_QP + 1 * 16 + c] = QA[1 * 16 + c];
    w[OFF_QP + 2 * 16 + c] = QBHA[0 * 16 + c];
    w[OFF_QP + 3 * 16 + c] = QBHA[1 * 16 + c];
  }
  float Rc[16];                                        // Rc = [[RA,0],[QB*PA,RB]]
  for (int i = 0; i < 16; ++i) Rc[i] = 0.f;
  Rc[0*4+0] = RA[0];   Rc[0*4+1] = RA[1];
  Rc[1*4+0] = RA[2];   Rc[1*4+1] = RA[3];
  Rc[2*4+0] = QBPA[0]; Rc[2*4+1] = QBPA[1]; Rc[2*4+2] = RB[0]; Rc[2*4+3] = RB[1];
  Rc[3*4+0] = QBPA[2]; Rc[3*4+1] = QBPA[3]; Rc[3*4+2] = RB[2]; Rc[3*4+3] = RB[3];
  for (int i = 0; i < 64; ++i) w[OFF_RP + i] = 0.f;
  for (int i = 0; i < 16; ++i) w[OFF_RP + i] = Rc[i];  // rows 0..3 of [16][4]

  float Rt[16] = {1,0,0,0, 0,1,0,0, 0,0,1,0, 0,0,0,1};
  float Tm[16];
  for (int n = 0; n < NBLK; ++n) {                     // Rc^NBLK
    mm(Rc, Rt, Tm, 4, 4, 4);
    for (int i = 0; i < 16; ++i) Rt[i] = Tm[i];
  }
  for (int i = 0; i < 16; ++i) w[OFF_RT + i] = Rt[i];
}

__global__ void setupK(const float* __restrict__ bp, const float* __restrict__ lp,
                       float* __restrict__ coef) {
  if (threadIdx.x != 0) return;
  buildStage(bp,     bp + 6, coef);                 // bandpass cascade
  buildStage(lp,     lp + 6, coef + COEF_STRIDE);   // lowpass cascade
}

// ---------------------------------------------------------------------------
// WMMA helpers (wave32, f32 16x16x4)
// ---------------------------------------------------------------------------
__device__ inline v8f wmma4(v2f a, v2f b, v8f c) {
  // D = A(16x4) * B(4x16) + C   — f32 tensor op
  return __builtin_amdgcn_wmma_f32_16x16x4_f32(false, a, false, b,
                                               (short)0, c, false, false);
}

// A operand chunk kk of a row-major [16][16] matrix: lane m = M, K = 4kk+j+kh
__device__ inline v2f loadA16(const float* M, int kk, int m, int kh) {
  v2f a;
  a[0] = M[m * 16 + kk * 4 + kh + 0];
  a[1] = M[m * 16 + kk * 4 + kh + 1];
  return a;
}
// A operand of a row-major [16][4] matrix
__device__ inline v2f loadA4(const float* M, int m, int kh) {
  v2f a;
  a[0] = M[m * 4 + kh + 0];
  a[1] = M[m * 4 + kh + 1];
  return a;
}

// D rows 0..3 (new states) -> B-layout state operand
__device__ inline void d2state(v8f d, bool low, v2f& s) {
  float u2 = __shfl_xor(d[2], 16, 32);
  float u3 = __shfl_xor(d[3], 16, 32);
  s[0] = low ? d[0] : u2;
  s[1] = low ? d[1] : u3;
}

// full 16x16 D (time x channel) -> four B-operand K-chunks, half-swap only
__device__ inline void d2bchunks(v8f w, bool low, v2f bw[4]) {
  float t0 = __shfl_xor(w[0], 16, 32), t1 = __shfl_xor(w[1], 16, 32);
  float t2 = __shfl_xor(w[2], 16, 32), t3 = __shfl_xor(w[3], 16, 32);
  float t4 = __shfl_xor(w[4], 16, 32), t5 = __shfl_xor(w[5], 16, 32);
  float t6 = __shfl_xor(w[6], 16, 32), t7 = __shfl_xor(w[7], 16, 32);
  bw[0][0] = low ? w[0] : t2;  bw[0][1] = low ? w[1] : t3;   // K = 0..3
  bw[1][0] = low ? w[4] : t6;  bw[1][1] = low ? w[5] : t7;   // K = 4..7
  bw[2][0] = low ? t0 : w[2];  bw[2][1] = low ? t1 : w[3];   // K = 8..11
  bw[3][0] = low ? t4 : w[6];  bw[3][1] = low ? t5 : w[7];   // K = 12..15
}

// ---------------------------------------------------------------------------
// Sweep kernel. PHASE 1: stage1 state-only -> V1.
//               PHASE 3: stage1 exact + square + stage2 state-only -> V2.
//               PHASE 5: both stages exact -> out.
// grid (NCT, NCHUNK) x 32 threads (one wave per workgroup).
// ---------------------------------------------------------------------------
template <int PHASE>
__global__ __launch_bounds__(32)
void sweepK(const float* __restrict__ x, const float* __restrict__ coef,
            const float* __restrict__ st1, const float* __restrict__ st2,
            float* __restrict__ outOrV) {
  const int  lane = threadIdx.x;
  const int  m    = lane & 15;
  const bool low  = lane < 16;
  const int  kh   = low ? 0 : 2;
  const int  ct   = blockIdx.x;
  const int  chunk= blockIdx.y;
  const int  bb   = ct >> 2;
  const int  c0   = (ct & 3) * 16;
  const float* xB = x + (size_t)bb * TLEN * CCH + c0;

  const float* C1 = coef;
  const float* C2 = coef + COEF_STRIDE;

  // hoisted A operands (loop invariant)
  v2f Qp1[4], Rp1, Hc1[4], Pc1;
  #pragma unroll
  for (int kk = 0; kk < 4; ++kk) Qp1[kk] = loadA16(C1 + OFF_QP, kk, m, kh);
  Rp1 = loadA4(C1 + OFF_RP, m, kh);
  if (PHASE >= 3) {
    #pragma unroll
    for (int kk = 0; kk < 4; ++kk) Hc1[kk] = loadA16(C1 + OFF_HC, kk, m, kh);
    Pc1 = loadA4(C1 + OFF_PC, m, kh);
  }
  v2f Qp2[4], Rp2, Hc2[4], Pc2;
  if (PHASE >= 3) {
    #pragma unroll
    for (int kk = 0; kk < 4; ++kk) Qp2[kk] = loadA16(C2 + OFF_QP, kk, m, kh);
    Rp2 = loadA4(C2 + OFF_RP, m, kh);
  }
  if (PHASE == 5) {
    #pragma unroll
    for (int kk = 0; kk < 4; ++kk) Hc2[kk] = loadA16(C2 + OFF_HC, kk, m, kh);
    Pc2 = loadA4(C2 + OFF_PC, m, kh);
  }

  // entering states (B layout: lane half holds states kh, kh+1)
  v2f s1 = {0.f, 0.f}, s2 = {0.f, 0.f};
  if (PHASE >= 3) {
    const float* S = st1 + ((size_t)chunk * NCT + ct) * 64;
    s1[0] = S[(kh + 0) * 16 + m];
    s1[1] = S[(kh + 1) * 16 + m];
  }
  if (PHASE == 5) {
    const float* S = st2 + ((size_t)chunk * NCT + ct) * 64;
    s2[0] = S[(kh + 0) * 16 + m];
    s2[1] = S[(kh + 1) * 16 + m];
  }

  const int tbase = chunk * LCH;
  for (int blk = 0; blk < NBLK; ++blk) {
    const int t0 = tbase + blk * 16;
    if (blk + 1 < NBLK)
      __builtin_prefetch(&xB[(t0 + 16 + kh) * CCH + m], 0, 0);

    // X block directly in B-operand layout: K-row = 4kk + j + kh
    v2f bx[4];
    #pragma unroll
    for (int kk = 0; kk < 4; ++kk) {
      bx[kk][0] = xB[(t0 + 4 * kk + kh + 0) * CCH + m];
      bx[kk][1] = xB[(t0 + 4 * kk + kh + 1) * CCH + m];
    }

    if (PHASE == 1) {
      v8f d = {};
      #pragma unroll
      for (int kk = 0; kk < 4; ++kk) d = wmma4(Qp1[kk], bx[kk], d);
      d = wmma4(Rp1, s1, d);
      d2state(d, low, s1);
    } else {
      // stage 1 exact: Y1 = Hc1*X + Pc1*S1 ; S1' = Qc1*X + Rc1*S1
      v8f y = {};
      #pragma unroll
      for (int kk = 0; kk < 4; ++kk) y = wmma4(Hc1[kk], bx[kk], y);
      y = wmma4(Pc1, s1, y);
      v8f ds = {};
      #pragma unroll
      for (int kk = 0; kk < 4; ++kk) ds = wmma4(Qp1[kk], bx[kk], ds);
      ds = wmma4(Rp1, s1, ds);
      d2state(ds, low, s1);

      // square (pointwise), reslice D layout -> B chunks for stage 2
      v8f w;
      #pragma unroll
      for (int i = 0; i < 8; ++i) w[i] = y[i] * y[i];
      v2f bw[4];
      d2bchunks(w, low, bw);

      if (PHASE == 3) {
        v8f d2 = {};
        #pragma unroll
        for (int kk = 0; kk < 4; ++kk) d2 = wmma4(Qp2[kk], bw[kk], d2);
        d2 = wmma4(Rp2, s2, d2);
        d2state(d2, low, s2);
      } else {  // PHASE 5
        v8f y2 = {};
        #pragma unroll
        for (int kk = 0; kk < 4; ++kk) y2 = wmma4(Hc2[kk], bw[kk], y2);
        y2 = wmma4(Pc2, s2, y2);
        v8f d2 = {};
        #pragma unroll
        for (int kk = 0; kk < 4; ++kk) d2 = wmma4(Qp2[kk], bw[kk], d2);
        d2 = wmma4(Rp2, s2, d2);
        d2state(d2, low, s2);

        // store: D layout row r -> time t0 + r + 8*half, channel c0 + m
        float* o = outOrV + (size_t)bb * TLEN * CCH + c0;
        const int tw = t0 + (low ? 0 : 8);
        #pragma unroll
        for (int r = 0; r < 8; ++r) o[(tw + r) * CCH + m] = y2[r];
      }
    }
  }

  if (PHASE == 1 || PHASE == 3) {   // zero-state exit states -> V
    float* V = outOrV + ((size_t)chunk * NCT + ct) * 64;
    v2f s = (PHASE == 1) ? s1 : s2;
    V[(kh + 0) * 16 + m] = s[0];
    V[(kh + 1) * 16 + m] = s[1];
  }
}

// ---------------------------------------------------------------------------
// Chunk-state scan: S_in[0] = 0 ; S_in[k+1] = Rtot * S_in[k] + V[k]
// one thread per (ctile, channel): 1024 threads, 32-step 4-vector scan.
// ---------------------------------------------------------------------------
__global__ __launch_bounds__(32)
void scanK(const float* __restrict__ V, const float* __restrict__ Rt,
           float* __restrict__ ST) {
  const int tid = blockIdx.x * 32 + threadIdx.x;   // 0..1023
  const int ct  = tid >> 4;
  const int ch  = tid & 15;
  float R[16];
  #pragma unroll
  for (int i = 0; i < 16; ++i) R[i] = Rt[i];
  float s0 = 0.f, s1 = 0.f, s2 = 0.f, s3 = 0.f;
  for (int k = 0; k < NCHUNK; ++k) {
    const size_t base = ((size_t)k * NCT + ct) * 64 + ch;
    ST[base +  0] = s0; ST[base + 16] = s1;
    ST[base + 32] = s2; ST[base + 48] = s3;
    const float v0 = V[base], v1 = V[base + 16], v2 = V[base + 32], v3 = V[base + 48];
    const float n0 = v0 + R[ 0]*s0 + R[ 1]*s1 + R[ 2]*s2 + R[ 3]*s3;
    const float n1 = v1 + R[ 4]*s0 + R[ 5]*s1 + R[ 6]*s2 + R[ 7]*s3;
    const float n2 = v2 + R[ 8]*s0 + R[ 9]*s1 + R[10]*s2 + R[11]*s3;
    const float n3 = v3 + R[12]*s0 + R[13]*s1 + R[14]*s2 + R[15]*s3;
    s0 = n0; s1 = n1; s2 = n2; s3 = n3;
  }
}

// ---------------------------------------------------------------------------
extern "C" void kernel_launch(void* const* d_in, const int* in_sizes, int n_in,
                              void* d_out, int out_size, void* d_ws, size_t ws_size,
                              hipStream_t stream) {
  (void)in_sizes; (void)n_in; (void)out_size; (void)ws_size;
  const float* x  = (const float*)d_in[0];
  const float* bp = (const float*)d_in[1];
  const float* lp = (const float*)d_in[2];
  float* out = (float*)d_out;
  float* ws  = (float*)d_ws;

  float* coef = ws;
  float* V1  = ws + OFF_V1;
  float* ST1 = ws + OFF_ST1;
  float* V2  = ws + OFF_V2;
  float* ST2 = ws + OFF_ST2;

  setupK<<<1, 32, 0, stream>>>(bp, lp, coef);

  dim3 grid(NCT, NCHUNK);
  sweepK<1><<<grid, 32, 0, stream>>>(x, coef, nullptr, nullptr, V1);
  scanK<<<32, 32, 0, stream>>>(V1, coef + 0 * COEF_STRIDE + OFF_RT, ST1);
  sweepK<3><<<grid, 32, 0, stream>>>(x, coef, ST1, nullptr, V2);
  scanK<<<32, 32, 0, stream>>>(V2, coef + 1 * COEF_STRIDE + OFF_RT, ST2);
  sweepK<5><<<grid, 32, 0, stream>>>(x, coef, ST1, ST2, out);
}